// ChannelAggregator_48661979464237
// MI455X (gfx1250) — compile-verified
//
#include <hip/hip_runtime.h>

// ---------------------------------------------------------------------------
// ChannelAggregator for MI455X (gfx1250, wave32)
//
// Bucket rows by channel (no float atomics), then per-channel gather-reduce
// with V_WMMA_F32_16X16X4_F32 using an all-ones A matrix:
//   D[m][n] = sum_k B[k][n] + C[m][n]   -> each WMMA folds 4 rows x 16 cols.
// B-matrix f32 4x16 layout (wave32): VGPR0 = rows K=0 (lanes 0-15) / K=2
// (lanes 16-31); VGPR1 = rows K=1 / K=3. D row M=0 = acc[0], lanes 0-15.
//
// Round-2 fixes: LDS stores pre-multiplied element offsets (row*128) so z is
// addressed with a 32-bit voffset (SADDR + scale_offset form, no 64-bit VALU
// chains); sidx is sentinel-padded so the index pair is one aligned
// ds_load_b64 and all clamping v_min ops vanish.
// ---------------------------------------------------------------------------

typedef __attribute__((ext_vector_type(2))) float v2f;
typedef __attribute__((ext_vector_type(8))) float v8f;

#define NUM_CH 4096
#define DIM    128
#define CHUNK  1024   // bucket indices staged per LDS pass (4 KB + pad)
#define PAD    12

// Workspace layout (bytes):
//   [0,   16K) : unsigned cnt[4096]
//   [16K, 32K) : int      first_idx[4096]
//   [32K, 48K) : unsigned offs[4096]
//   [48K, 64K) : unsigned cursor[4096]
//   [64K, 64K + 4*B) : int bucket[B]
// Total for B=524288: ~2.06 MB.

__global__ void ca_init(unsigned* __restrict__ cnt, int* __restrict__ first_idx, int B) {
    int i = blockIdx.x * blockDim.x + threadIdx.x;
    if (i < NUM_CH) {
        cnt[i] = 0u;
        first_idx[i] = B;   // matches reference fill value
    }
}

__global__ void ca_hist(const int* __restrict__ ch, unsigned* __restrict__ cnt,
                        int* __restrict__ first_idx, int B) {
    int i = blockIdx.x * blockDim.x + threadIdx.x;
    if (i < B) {
        int c = ch[i];
        atomicAdd(&cnt[c], 1u);
        atomicMin(&first_idx[c], i);
    }
}

// Exclusive scan over 4096 counts: one block of 1024 threads, 4 elems/thread.
__global__ void __launch_bounds__(1024)
ca_scan(const unsigned* __restrict__ cnt, unsigned* __restrict__ offs,
        unsigned* __restrict__ cursor) {
    __shared__ unsigned part[1024];
    int t = threadIdx.x;
    unsigned v0 = cnt[4 * t + 0];
    unsigned v1 = cnt[4 * t + 1];
    unsigned v2 = cnt[4 * t + 2];
    unsigned v3 = cnt[4 * t + 3];
    part[t] = v0 + v1 + v2 + v3;
    __syncthreads();
    for (int off = 1; off < 1024; off <<= 1) {
        unsigned add = (t >= off) ? part[t - off] : 0u;
        __syncthreads();
        part[t] += add;
        __syncthreads();
    }
    unsigned base = (t == 0) ? 0u : part[t - 1];
    unsigned o0 = base;
    unsigned o1 = o0 + v0;
    unsigned o2 = o1 + v1;
    unsigned o3 = o2 + v2;
    offs[4 * t + 0] = o0;  cursor[4 * t + 0] = o0;
    offs[4 * t + 1] = o1;  cursor[4 * t + 1] = o1;
    offs[4 * t + 2] = o2;  cursor[4 * t + 2] = o2;
    offs[4 * t + 3] = o3;  cursor[4 * t + 3] = o3;
}

__global__ void ca_scatter(const int* __restrict__ ch, unsigned* __restrict__ cursor,
                           int* __restrict__ bucket, int B) {
    int i = blockIdx.x * blockDim.x + threadIdx.x;
    if (i < B) {
        int c = ch[i];
        unsigned p = atomicAdd(&cursor[c], 1u);  // absolute (cursor==offs at start)
        bucket[p] = i;
    }
}

// One 256-thread block (8 waves) per channel; wave w owns columns [16w,16w+16).
__global__ void __launch_bounds__(256)
ca_reduce(const float* __restrict__ z, const int* __restrict__ y,
          const unsigned* __restrict__ cnt, const unsigned* __restrict__ offs,
          const int* __restrict__ first_idx, const int* __restrict__ bucket,
          float* __restrict__ out, int B) {
    __shared__ int sidx[CHUNK + PAD];   // pre-multiplied element offsets (row*128)

    const int c    = blockIdx.x;
    const int tid  = threadIdx.x;
    const int wave = tid >> 5;
    const int lane = tid & 31;
    const int nn   = lane & 15;      // column within the 16-wide tile
    const int kh   = lane >> 4;      // 0: K rows {0,1}; 1: K rows {2,3}
    const int col  = wave * 16 + nn; // this lane's z column

    const int count     = (int)cnt[c];
    const unsigned base = offs[c];

    v2f a;
    a.x = 1.0f;
    a.y = 1.0f;
    v8f acc = {};

    for (int cb = 0; cb < count; cb += CHUNK) {
        const int n = (count - cb < CHUNK) ? (count - cb) : CHUNK;

        // Cooperative coalesced staging: store row*DIM (31-bit safe).
        for (int j = tid; j < n; j += 256)
            sidx[j] = bucket[base + cb + j] * DIM;
        // Sentinel pad: offset 0 is always a valid z address; padded lanes
        // are zeroed by cndmask, so the value is irrelevant.
        if (tid < PAD)
            sidx[n + tid] = 0;
        __syncthreads();

        const int tiles = (n + 3) >> 2;

        // ---- tile loader: one aligned ds_load_b64 for the index pair,
        //      32-bit voffset z loads (SADDR + scale_offset), cndmask tail.
        auto load_tile = [&](int t, float& obx, float& oby) {
            int rx = 4 * t + 2 * kh;                     // even, < n + 4
            int2 pr = *(const int2*)(sidx + rx);         // ds_load_b64
            float vx = z[(unsigned)(pr.x + col)];
            float vy = z[(unsigned)(pr.y + col)];
            obx = (rx < n)     ? vx : 0.0f;
            oby = (rx + 1 < n) ? vy : 0.0f;
        };

        float bx, by;
        load_tile(0, bx, by);

        for (int t = 0; t < tiles; ++t) {
            // Double buffer: issue tile t+1's loads before the WMMA of tile t.
            int tn = (t + 1 < tiles) ? (t + 1) : t;
            float nbx, nby;
            load_tile(tn, nbx, nby);

            // GL2 prefetch two tiles ahead (global_prefetch_b8, SE scope).
            int rp = 4 * t + 8 + 2 * kh;                 // < n + PAD
            __builtin_prefetch(&z[(unsigned)(sidx[rp] + col)], 0, 3);

            v2f b;
            b.x = bx;
            b.y = by;
            acc = __builtin_amdgcn_wmma_f32_16x16x4_f32(
                /*neg_a=*/false, a, /*neg_b=*/false, b,
                /*c_mod=*/(short)0, acc, /*reuse_a=*/false, /*reuse_b=*/false);

            bx = nbx;
            by = nby;
        }
        __syncthreads();   // protect sidx before next chunk restage
    }

    // D-matrix row M=0 lives in acc[0], lanes 0-15 (N = 0..15).
    const float inv = 1.0f / fmaxf((float)count, 1.0f);
    if (lane < 16)
        out[c * DIM + wave * 16 + lane] = acc[0] * inv;

    if (tid == 0) {
        int fi = first_idx[c];
        if (fi > B - 1) fi = B - 1;   // empty channel -> y[B-1] (matches reference)
        out[NUM_CH * DIM + c] = (float)y[fi];
    }
}

extern "C" void kernel_launch(void* const* d_in, const int* in_sizes, int n_in,
                              void* d_out, int out_size, void* d_ws, size_t ws_size,
                              hipStream_t stream) {
    const float* z  = (const float*)d_in[0];
    const int*   y  = (const int*)d_in[1];
    const int*   ch = (const int*)d_in[2];
    const int B = in_sizes[1];            // y has one entry per row
    float* out = (float*)d_out;

    unsigned char* ws = (unsigned char*)d_ws;
    unsigned* cnt       = (unsigned*)(ws);
    int*      first_idx = (int*)     (ws + NUM_CH * 4);
    unsigned* offs      = (unsigned*)(ws + NUM_CH * 8);
    unsigned* cursor    = (unsigned*)(ws + NUM_CH * 12);
    int*      bucket    = (int*)     (ws + NUM_CH * 16);

    ca_init   <<<(NUM_CH + 255) / 256, 256, 0, stream>>>(cnt, first_idx, B);
    ca_hist   <<<(B + 255) / 256,      256, 0, stream>>>(ch, cnt, first_idx, B);
    ca_scan   <<<1,                   1024, 0, stream>>>(cnt, offs, cursor);
    ca_scatter<<<(B + 255) / 256,      256, 0, stream>>>(ch, cursor, bucket, B);
    ca_reduce <<<NUM_CH,               256, 0, stream>>>(z, y, cnt, offs, first_idx,
                                                         bucket, out, B);
}